// Decoder_72284299592327
// MI455X (gfx1250) — compile-verified
//
#include <hip/hip_runtime.h>
#include <hip/hip_bf16.h>
#include <math.h>

// ---------------- problem constants ----------------
constexpr int B     = 64;
constexpr int TX    = 1024;
constexpr int ADIM  = 1024;
constexpr int UNITS = 1024;
constexpr int VOCAB = 32000;
constexpr int EDIM  = 512;
constexpr int XDIM  = ADIM + EDIM;   // 1536
constexpr int ZDIM  = 4 * UNITS;     // 4096

// output layout (floats): y_pred, context, alpha, h_new, c_new
constexpr long OUT_Y     = 0;
constexpr long OUT_CTX   = (long)B * VOCAB;            // 2,048,000
constexpr long OUT_ALPHA = OUT_CTX + (long)B * ADIM;   // +65,536
constexpr long OUT_H     = OUT_ALPHA + (long)B * TX;   // +65,536
constexpr long OUT_C     = OUT_H + (long)B * UNITS;    // +65,536

// workspace layout (floats)
constexpr long WS_PARTIAL = 0;                                // B*8*ADIM = 524,288
constexpr long WS_XIN     = WS_PARTIAL + (long)B * 8 * ADIM;  // B*XDIM = 98,304
constexpr long WS_Z       = WS_XIN + (long)B * XDIM;          // B*ZDIM = 262,144

// ---------------- WMMA fp32 fragments ----------------
typedef __attribute__((ext_vector_type(2))) float v2f;
typedef __attribute__((ext_vector_type(8))) float v8f;

// A-matrix 16x4 f32 (ISA 7.12.2): lanes 0-15 hold M=0..15; VGPR0 = K {0 | 2},
// VGPR1 = K {1 | 3} for lane<16 | lane>=16.
__device__ __forceinline__ v2f load_a_frag(const float* __restrict__ A, int lda,
                                           int row0, int k0, int lane) {
  const int m  = row0 + (lane & 15);
  const int kk = k0 + ((lane >> 4) << 1);
  const float* p = A + (size_t)m * lda + kk;
  v2f r; r.x = p[0]; r.y = p[1];
  return r;
}

// B-matrix 4x16 f32: rows striped across lanes within a VGPR (mirrors C/D
// layout): VGPR0 = row K {0 | 2}, VGPR1 = row K {1 | 3}; N = lane % 16.
__device__ __forceinline__ v2f load_b_frag(const float* __restrict__ W, int ldb,
                                           int k0, int col0, int lane) {
  const int n  = col0 + (lane & 15);
  const int kk = k0 + ((lane >> 4) << 1);
  v2f r;
  r.x = W[(size_t)kk * ldb + n];
  r.y = W[(size_t)(kk + 1) * ldb + n];
  return r;
}

// Branch-free prefetch of the 4 K-rows we will consume PF_K rows from now.
// lane&3 spreads lanes over rows k..k+3 (each 16KB-strided line hit by 8 lanes,
// merged by the coalescer). kpf is uniform -> scalar min, no control flow.
// Lowers to gfx1250 global_prefetch_b8 (GL2 fill, no WGP pollution).
constexpr int PF_K = 64;
__device__ __forceinline__ void prefetch_b_rows(const float* __restrict__ W, int ldb,
                                                int k0, int K, int col0, int lane) {
  int kpf = k0 + PF_K;
  kpf = (kpf < K - 4) ? kpf : (K - 4);
  __builtin_prefetch(&W[(size_t)(kpf + (lane & 3)) * ldb + col0], 0, 0);
}

#define WMMA_F32(a, b, c) \
  __builtin_amdgcn_wmma_f32_16x16x4_f32(false, (a), false, (b), (short)0, (c), false, false)

// ---------------- kernel 1: partial sum of a over t-chunks ----------------
// grid = B*8 blocks, 256 threads; each thread owns 4 consecutive d (float4).
// partial[(b*8+tc)*ADIM + d] = sum over 128 t of a[b, tc*128+t, d].
__global__ void partial_sum_kernel(const float* __restrict__ a, float* __restrict__ partial) {
  const int b  = blockIdx.x >> 3;
  const int tc = blockIdx.x & 7;
  const int d4 = threadIdx.x;                       // float4 index, 0..255
  const float4* __restrict__ a4 =
      (const float4*)(a + ((size_t)b * TX + (size_t)tc * 128) * ADIM) + d4;
  float4 acc = make_float4(0.f, 0.f, 0.f, 0.f);
#pragma unroll 8
  for (int t = 0; t < 128; ++t) {
    const float4 v = a4[(size_t)t * (ADIM / 4)];    // global_load_b128, coalesced
    acc.x += v.x; acc.y += v.y; acc.z += v.z; acc.w += v.w;
  }
  ((float4*)(partial + (size_t)blockIdx.x * ADIM))[d4] = acc;
}

// ---------------- kernel 2: finalize context -> output + xin ----------------
// B*ADIM/4 threads; each handles 4 consecutive d.
__global__ void finalize_ctx_kernel(const float* __restrict__ partial,
                                    float* __restrict__ ctx_out,
                                    float* __restrict__ xin) {
  const int idx4 = blockIdx.x * blockDim.x + threadIdx.x;  // 0..B*ADIM/4-1
  const int b  = idx4 >> 8;                                // ADIM/4 = 256 per row
  const int d4 = idx4 & 255;
  float4 acc = make_float4(0.f, 0.f, 0.f, 0.f);
#pragma unroll
  for (int tc = 0; tc < 8; ++tc) {
    const float4 v = ((const float4*)(partial + (size_t)(b * 8 + tc) * ADIM))[d4];
    acc.x += v.x; acc.y += v.y; acc.z += v.z; acc.w += v.w;
  }
  ((float4*)(ctx_out + (size_t)b * ADIM))[d4] = acc;
  ((float4*)(xin + (size_t)b * XDIM))[d4]     = acc;
}

// ---------------- kernel 3: embedding gather -> xin ----------------
__global__ void embed_kernel(const int* __restrict__ X, const float* __restrict__ emb,
                             float* __restrict__ xin) {
  const int idx = blockIdx.x * blockDim.x + threadIdx.x;  // 0..B*EDIM-1
  const int b = idx / EDIM;
  const int e = idx - b * EDIM;
  xin[(size_t)b * XDIM + ADIM + e] = emb[(size_t)X[b] * EDIM + e];
}

// ---------------- kernel 4: alpha = softmax over size-1 axis == 1.0 ----------------
__global__ void fill_alpha_kernel(float* __restrict__ alpha) {
  alpha[blockIdx.x * blockDim.x + threadIdx.x] = 1.0f;
}

// ---------------- kernel 5: z = xin@Wx + h@Wh + bl (WMMA f32) ----------------
// One wave per 16-col tile of z (256 tiles); 4 M-tiles per wave (M=64).
__global__ void gemm_z_kernel(const float* __restrict__ xin, const float* __restrict__ h,
                              const float* __restrict__ Wx, const float* __restrict__ Wh,
                              const float* __restrict__ bl, float* __restrict__ z) {
  const int wave = threadIdx.x >> 5;
  const int lane = threadIdx.x & 31;
  const int nt   = blockIdx.x * 8 + wave;   // 0..255
  const int col0 = nt * 16;

  v8f acc[4] = {};

  // xin [64 x 1536] @ Wx [1536 x 4096]
  for (int k0 = 0; k0 < XDIM; k0 += 4) {
    prefetch_b_rows(Wx, ZDIM, k0, XDIM, col0, lane);
    const v2f bf = load_b_frag(Wx, ZDIM, k0, col0, lane);
#pragma unroll
    for (int mt = 0; mt < 4; ++mt) {
      const v2f af = load_a_frag(xin, XDIM, mt * 16, k0, lane);
      acc[mt] = WMMA_F32(af, bf, acc[mt]);
    }
  }
  // h [64 x 1024] @ Wh [1024 x 4096]
  for (int k0 = 0; k0 < UNITS; k0 += 4) {
    prefetch_b_rows(Wh, ZDIM, k0, UNITS, col0, lane);
    const v2f bf = load_b_frag(Wh, ZDIM, k0, col0, lane);
#pragma unroll
    for (int mt = 0; mt < 4; ++mt) {
      const v2f af = load_a_frag(h, UNITS, mt * 16, k0, lane);
      acc[mt] = WMMA_F32(af, bf, acc[mt]);
    }
  }

  // C/D layout: VGPR r -> row r (lanes 0-15) / row r+8 (lanes 16-31), N = lane%16
  const int col    = col0 + (lane & 15);
  const int rowAdd = (lane < 16) ? 0 : 8;
  const float bias = bl[col];
#pragma unroll
  for (int mt = 0; mt < 4; ++mt)
#pragma unroll
    for (int r = 0; r < 8; ++r)
      z[(size_t)(mt * 16 + r + rowAdd) * ZDIM + col] = acc[mt][r] + bias;
}

// ---------------- kernel 6: LSTM gates (Keras order i,f,g,o) ----------------
__device__ __forceinline__ float sigf(float x) { return 1.0f / (1.0f + expf(-x)); }

__global__ void lstm_elem_kernel(const float* __restrict__ z, const float* __restrict__ c,
                                 float* __restrict__ h_new, float* __restrict__ c_new) {
  const int idx = blockIdx.x * blockDim.x + threadIdx.x;  // 0..B*UNITS-1
  const int b = idx >> 10;
  const int u = idx & 1023;
  const float* zb = z + (size_t)b * ZDIM;
  const float gi = zb[u];
  const float gf = zb[u + UNITS];
  const float gg = zb[u + 2 * UNITS];
  const float go = zb[u + 3 * UNITS];
  const float cn = sigf(gf) * c[idx] + sigf(gi) * tanhf(gg);
  c_new[idx] = cn;
  h_new[idx] = sigf(go) * tanhf(cn);
}

// ---------------- kernel 7: logits = h_new @ Wv + bv (WMMA f32) ----------------
// 2000 column tiles; one wave each, 4 M-tiles per wave.
__global__ void gemm_vocab_kernel(const float* __restrict__ hn, const float* __restrict__ Wv,
                                  const float* __restrict__ bv, float* __restrict__ logits) {
  const int wave = threadIdx.x >> 5;
  const int lane = threadIdx.x & 31;
  const int nt   = blockIdx.x * 8 + wave;   // 0..1999
  const int col0 = nt * 16;

  v8f acc[4] = {};
  for (int k0 = 0; k0 < UNITS; k0 += 4) {
    prefetch_b_rows(Wv, VOCAB, k0, UNITS, col0, lane);
    const v2f bf = load_b_frag(Wv, VOCAB, k0, col0, lane);
#pragma unroll
    for (int mt = 0; mt < 4; ++mt) {
      const v2f af = load_a_frag(hn, UNITS, mt * 16, k0, lane);
      acc[mt] = WMMA_F32(af, bf, acc[mt]);
    }
  }

  const int col    = col0 + (lane & 15);
  const int rowAdd = (lane < 16) ? 0 : 8;
  const float bias = bv[col];
#pragma unroll
  for (int mt = 0; mt < 4; ++mt)
#pragma unroll
    for (int r = 0; r < 8; ++r)
      logits[(size_t)(mt * 16 + r + rowAdd) * VOCAB + col] = acc[mt][r] + bias;
}

// ---------------- kernel 8: in-place row softmax over VOCAB ----------------
__global__ void softmax_rows_kernel(float* __restrict__ y) {
  __shared__ float sdata[256];
  const int b   = blockIdx.x;
  const int tid = threadIdx.x;
  float* row = y + (size_t)b * VOCAB;

  float m = -3.4e38f;
  for (int i = tid; i < VOCAB; i += 256) m = fmaxf(m, row[i]);
  sdata[tid] = m; __syncthreads();
  for (int s = 128; s > 0; s >>= 1) {
    if (tid < s) sdata[tid] = fmaxf(sdata[tid], sdata[tid + s]);
    __syncthreads();
  }
  m = sdata[0]; __syncthreads();

  float sum = 0.f;
  for (int i = tid; i < VOCAB; i += 256) sum += expf(row[i] - m);
  sdata[tid] = sum; __syncthreads();
  for (int s = 128; s > 0; s >>= 1) {
    if (tid < s) sdata[tid] += sdata[tid + s];
    __syncthreads();
  }
  const float inv = 1.0f / sdata[0]; __syncthreads();

  for (int i = tid; i < VOCAB; i += 256) row[i] = expf(row[i] - m) * inv;
}

// ---------------- launcher ----------------
extern "C" void kernel_launch(void* const* d_in, const int* in_sizes, int n_in,
                              void* d_out, int out_size, void* d_ws, size_t ws_size,
                              hipStream_t stream) {
  const int*   X   = (const int*)  d_in[0];
  const float* a   = (const float*)d_in[1];
  const float* h   = (const float*)d_in[2];
  const float* c   = (const float*)d_in[3];
  const float* emb = (const float*)d_in[4];
  // d_in[5..10] = W1,b1,W2,b2,We,be : mathematically irrelevant (alpha == 1)
  const float* Wx  = (const float*)d_in[11];
  const float* Wh  = (const float*)d_in[12];
  const float* bl  = (const float*)d_in[13];
  const float* Wv  = (const float*)d_in[14];
  const float* bv  = (const float*)d_in[15];

  float* out = (float*)d_out;
  float* ws  = (float*)d_ws;

  float* partial = ws + WS_PARTIAL;
  float* xin     = ws + WS_XIN;
  float* z       = ws + WS_Z;

  // context = sum_t a[b,t,:] (alpha == 1 exactly: softmax over size-1 axis)
  partial_sum_kernel<<<B * 8, 256, 0, stream>>>(a, partial);
  finalize_ctx_kernel<<<(B * ADIM / 4) / 256, 256, 0, stream>>>(partial, out + OUT_CTX, xin);
  embed_kernel<<<(B * EDIM) / 256, 256, 0, stream>>>(X, emb, xin);
  fill_alpha_kernel<<<(B * TX) / 256, 256, 0, stream>>>(out + OUT_ALPHA);

  // z = xin@Wx + h@Wh + bl  (256 column tiles, 8 waves/block)
  gemm_z_kernel<<<32, 256, 0, stream>>>(xin, h, Wx, Wh, bl, z);
  lstm_elem_kernel<<<(B * UNITS) / 256, 256, 0, stream>>>(z, c, out + OUT_H, out + OUT_C);

  // logits into y region, then in-place softmax
  gemm_vocab_kernel<<<VOCAB / 16 / 8, 256, 0, stream>>>(out + OUT_H, Wv, bv, out + OUT_Y);
  softmax_rows_kernel<<<B, 256, 0, stream>>>(out + OUT_Y);
}